// NeuralDictionary_19430432047757
// MI455X (gfx1250) — compile-verified
//
#include <hip/hip_runtime.h>
#include <hip/hip_bf16.h>
#include <math.h>

typedef __attribute__((ext_vector_type(2))) float v2f;
typedef __attribute__((ext_vector_type(8))) float v8f;

#define NB   1024
#define TPB  128
#define WPB  (TPB / 32)
#define ROWPAD 68          // floats per LDS row: 64 data + 4 pad (272 B, 16B-aligned)

// ---------------------------------------------------------------------------
// Pass 1: async-double-buffered keys -> online softmax -> WMMA f32 16x16x4
// weighted-value accumulation. Each wave owns a private state and grid-strides
// over groups of 32 rows.
// ---------------------------------------------------------------------------
__global__ __launch_bounds__(TPB) void nd_pass1(
    const float* __restrict__ query,
    const float* __restrict__ keys,
    const float* __restrict__ values,
    float* __restrict__ ws, int C)
{
    __shared__ __align__(16) float kbuf[WPB][2][32 * ROWPAD]; // async staging
    __shared__ __align__(16) float swts[WPB][32];             // per-wave weights
    __shared__ float sm[WPB], sl[WPB], sbv[WPB];
    __shared__ int   sbi[WPB];
    __shared__ float sout[WPB][64];

    const int lane   = threadIdx.x & 31;
    const int wave   = threadIdx.x >> 5;
    const int gwave  = blockIdx.x * WPB + wave;
    const int nwaves = gridDim.x * WPB;
    const int ngroups = (C + 31) >> 5;

    // Query resident in registers.
    float q[64];
#pragma unroll
    for (int j = 0; j < 64; ++j) q[j] = query[j];

    float m    = -__builtin_inff();
    float lsum = 0.f;
    v8f acc0 = {}, acc1 = {}, acc2 = {}, acc3 = {};
    float bestv = -__builtin_inff();
    int   besti = 0;

    const int delta = (lane < 16) ? 0 : 2;  // K half-split across lane halves
    const int bcol  = lane & 15;            // N index for B/C/D layouts

    // Per-lane LDS byte offsets of this lane's destination row in each buffer.
    const unsigned lds0 = (unsigned)(size_t)&kbuf[wave][0][lane * ROWPAD];
    const unsigned lds1 = (unsigned)(size_t)&kbuf[wave][1][lane * ROWPAD];

    // Async copy of one 32x64 keys tile: lane l moves row (base+l) in 16B
    // chunks; the instruction IOFFSET advances global and LDS address together.
    auto prefetch_group = [&](int g, unsigned ldsdst) {
        long row = (long)g * 32 + lane;
        long rc  = row < (long)C ? row : (long)C - 1;   // clamp: weight==0 there
        const float* gp = keys + (size_t)rc * 64;
#pragma unroll
        for (int j = 0; j < 16; ++j)
            asm volatile("global_load_async_to_lds_b128 %0, %1, off offset:%2"
                         :: "v"(ldsdst), "v"(gp), "i"(16 * j) : "memory");
    };

    if (gwave < ngroups) prefetch_group(gwave, lds0);
    int cur = 0;

    for (int g = gwave; g < ngroups; g += nwaves) {
        // Kick off next tile into the other buffer before consuming this one.
        const int nxt = (g + nwaves < ngroups) ? (g + nwaves) : g;
        prefetch_group(nxt, cur ? lds0 : lds1);

        // 16 newest async ops may be outstanding; current tile must be done.
        asm volatile("s_wait_asynccnt 0x10" ::: "memory");

        // ---- score: dot(keys[row], query) from LDS (ds_load_b128 x16) ----
        const float4* krow = (const float4*)&kbuf[wave][cur][lane * ROWPAD];
        float d = 0.f;
#pragma unroll
        for (int j = 0; j < 16; ++j) {
            float4 kv = krow[j];
            d = fmaf(kv.x, q[4 * j + 0], d);
            d = fmaf(kv.y, q[4 * j + 1], d);
            d = fmaf(kv.z, q[4 * j + 2], d);
            d = fmaf(kv.w, q[4 * j + 3], d);
        }
        cur ^= 1;

        const int base = g << 5;
        const int row  = base + lane;
        float s;
        if (row < C) {
            s = d;
            if (s > bestv) { bestv = s; besti = row; }
        } else {
            s = -__builtin_inff();
        }

        // ---- online softmax: wave max, rescale running state ----
        float gm = s;
#pragma unroll
        for (int off = 16; off >= 1; off >>= 1)
            gm = fmaxf(gm, __shfl_xor(gm, off, 32));
        const float mn = fmaxf(m, gm);
        const float r  = __expf(m - mn);   // first iter: exp(-inf) == 0
        m = mn;
        lsum *= r;
        acc0 *= r; acc1 *= r; acc2 *= r; acc3 *= r;

        const float w = (row < C) ? __expf(s - mn) : 0.f;
        lsum += w;

        // ---- weights to LDS; A operands come back as aligned v2f pairs ----
        swts[wave][lane] = w;
        asm volatile("s_wait_dscnt 0x0" ::: "memory");

        // ---- weighted values via WMMA f32 16x16x4, 8 K-steps of 4 rows ----
        if (base + 32 <= C) {
            // Fast path: one base pointer, all loads use immediate offsets.
            const float* vp = values + ((size_t)base + delta) * 64 + bcol;
#pragma unroll
            for (int k = 0; k < 8; ++k) {
                const v2f A = *(const v2f*)&swts[wave][4 * k + delta];
                const float* p = vp + k * 256;
                const v2f B0 = { p[ 0], p[64 +  0] };
                const v2f B1 = { p[16], p[64 + 16] };
                const v2f B2 = { p[32], p[64 + 32] };
                const v2f B3 = { p[48], p[64 + 48] };
                acc0 = __builtin_amdgcn_wmma_f32_16x16x4_f32(false, A, false, B0, (short)0, acc0, false, false);
                acc1 = __builtin_amdgcn_wmma_f32_16x16x4_f32(false, A, false, B1, (short)0, acc1, false, false);
                acc2 = __builtin_amdgcn_wmma_f32_16x16x4_f32(false, A, false, B2, (short)0, acc2, false, false);
                acc3 = __builtin_amdgcn_wmma_f32_16x16x4_f32(false, A, false, B3, (short)0, acc3, false, false);
            }
        } else {
            // Tail path: clamp row indices; clamped rows carry weight 0.
#pragma unroll
            for (int k = 0; k < 8; ++k) {
                const v2f A = *(const v2f*)&swts[wave][4 * k + delta];
                int r0  = base + 4 * k + delta;
                int rc0 = (r0     < C) ? r0     : (C - 1);
                int rc1 = (r0 + 1 < C) ? r0 + 1 : (C - 1);
                const float* v0 = values + (size_t)rc0 * 64 + bcol;
                const float* v1 = values + (size_t)rc1 * 64 + bcol;
                const v2f B0 = { v0[ 0], v1[ 0] };
                const v2f B1 = { v0[16], v1[16] };
                const v2f B2 = { v0[32], v1[32] };
                const v2f B3 = { v0[48], v1[48] };
                acc0 = __builtin_amdgcn_wmma_f32_16x16x4_f32(false, A, false, B0, (short)0, acc0, false, false);
                acc1 = __builtin_amdgcn_wmma_f32_16x16x4_f32(false, A, false, B1, (short)0, acc1, false, false);
                acc2 = __builtin_amdgcn_wmma_f32_16x16x4_f32(false, A, false, B2, (short)0, acc2, false, false);
                acc3 = __builtin_amdgcn_wmma_f32_16x16x4_f32(false, A, false, B3, (short)0, acc3, false, false);
            }
        }
    }

    // ---- wave-level reductions ----
#pragma unroll
    for (int off = 16; off >= 1; off >>= 1)
        lsum += __shfl_xor(lsum, off, 32);
#pragma unroll
    for (int off = 16; off >= 1; off >>= 1) {
        float ov = __shfl_xor(bestv, off, 32);
        int   oi = __shfl_xor(besti, off, 32);
        if (ov > bestv || (ov == bestv && oi < besti)) { bestv = ov; besti = oi; }
    }

    if (lane == 0) { sm[wave] = m; sl[wave] = lsum; sbv[wave] = bestv; sbi[wave] = besti; }
    if (lane < 16) {
        // all 16 D rows identical (broadcast A); VGPR0, lanes 0..15 = cols N
        sout[wave][ 0 + lane] = acc0[0];
        sout[wave][16 + lane] = acc1[0];
        sout[wave][32 + lane] = acc2[0];
        sout[wave][48 + lane] = acc3[0];
    }
    __syncthreads();

    const int tid = threadIdx.x;
    if (tid < 64) {
        float M = sm[0];
#pragma unroll
        for (int w2 = 1; w2 < WPB; ++w2) M = fmaxf(M, sm[w2]);
        float O = 0.f, L = 0.f, bv = -__builtin_inff();
        int bi = 0;
#pragma unroll
        for (int w2 = 0; w2 < WPB; ++w2) {
            const float f = __expf(sm[w2] - M);
            O += sout[w2][tid] * f;
            L += sl[w2] * f;
            if (sbv[w2] > bv) { bv = sbv[w2]; bi = sbi[w2]; }
        }
        float* ws_m  = ws;
        float* ws_l  = ws + gridDim.x;
        float* ws_bv = ws + 2 * (size_t)gridDim.x;
        int*   ws_bi = (int*)(ws + 3 * (size_t)gridDim.x);
        float* ws_o  = ws + 4 * (size_t)gridDim.x;

        ws_o[(size_t)blockIdx.x * 64 + tid] = O;
        if (tid == 0) {
            ws_m[blockIdx.x]  = M;
            ws_l[blockIdx.x]  = L;
            ws_bv[blockIdx.x] = bv;
            ws_bi[blockIdx.x] = bi;
        }
    }
}

// ---------------------------------------------------------------------------
// Pass 2: combine NB block partials (log-sum-exp merge), finalize out + argmax.
// ---------------------------------------------------------------------------
__global__ __launch_bounds__(64) void nd_pass2(
    const float* __restrict__ ws, float* __restrict__ out, int nb, int out_size)
{
    const int tid = threadIdx.x;
    if (tid >= 64) return;

    const float* ws_m  = ws;
    const float* ws_l  = ws + nb;
    const float* ws_bv = ws + 2 * (size_t)nb;
    const int*   ws_bi = (const int*)(ws + 3 * (size_t)nb);
    const float* ws_o  = ws + 4 * (size_t)nb;

    float M = -__builtin_inff();
    for (int b = 0; b < nb; ++b) M = fmaxf(M, ws_m[b]);

    float O = 0.f, L = 0.f, bv = -__builtin_inff();
    int bi = 0;
    for (int b = 0; b < nb; ++b) {
        const float f = __expf(ws_m[b] - M);
        O += ws_o[(size_t)b * 64 + tid] * f;
        L += ws_l[b] * f;
        const float v = ws_bv[b];
        if (v > bv) { bv = v; bi = ws_bi[b]; }
    }
    out[tid] = O / L;
    if (tid == 0 && out_size > 64) out[64] = (float)bi;
}

extern "C" void kernel_launch(void* const* d_in, const int* in_sizes, int n_in,
                              void* d_out, int out_size, void* d_ws, size_t ws_size,
                              hipStream_t stream) {
    const float* query  = (const float*)d_in[0];
    const float* keys   = (const float*)d_in[1];
    const float* values = (const float*)d_in[2];
    const int C = in_sizes[1] / 64;   // capacity (rows of keys)

    float* ws = (float*)d_ws;         // [m|l|bv|bi : NB each][out : NB*64]

    nd_pass1<<<NB, TPB, 0, stream>>>(query, keys, values, ws, C);
    nd_pass2<<<1, 64, 0, stream>>>(ws, (float*)d_out, NB, out_size);
}